// smooth_network_31971736552075
// MI455X (gfx1250) — compile-verified
//
#include <hip/hip_runtime.h>
#include <math.h>

typedef float v2f __attribute__((ext_vector_type(2)));
typedef float v8f __attribute__((ext_vector_type(8)));

#define NSAMP 2048
#define LTOK  1024
#define CCH   60     // channels
#define CPAD  64     // padded
#define KCL   8      // clusters

// ---------------------------------------------------------------------------
// Fused kernel: per-sample stats -> label -> XA[n] = X[n] @ A[label].
// One block (8 waves) per sample.
//  Phase 1: mean/std(ddof=1) reduction over X[n] (coalesced over channels),
//           nearest-centroid argmin -> s_label (+ labels[n] for the BW gather).
//           This pass warms L2 with the block's 245 KB slice of X.
//  Phase 2: WMMA f32 16x16x4 matmul. B-operand (A[label]) stationary in
//           120 VGPRs per wave; X streamed as A-operand fragments via
//           global_load_b64 — these re-reads hit L2 from phase 1.
// ---------------------------------------------------------------------------
__global__ __launch_bounds__(256) void fused_stats_xa_kernel(
    const float* __restrict__ X,          // [N, L, C]
    const float* __restrict__ centroids,  // [K, 2C]
    const float* __restrict__ A_all,      // [K, C, C]
    int* __restrict__ labels,             // [N]
    float* __restrict__ out)              // [N, L, C]
{
    __shared__ float s_sum[4][64];
    __shared__ float s_sq[4][64];
    __shared__ float s_stats[2 * CCH];
    __shared__ float s_d2[KCL];
    __shared__ int   s_label;

    const int n = blockIdx.x;
    const float* Xn = X + (size_t)n * LTOK * CCH;
    float* On = out + (size_t)n * LTOK * CCH;

    // ---------------- Phase 1: channel stats + kmeans label ----------------
    {
        const int c  = threadIdx.x & 63;   // channel lane (coalesced over c)
        const int rg = threadIdx.x >> 6;   // row group 0..3

        float sum = 0.0f, sq = 0.0f;
        if (c < CCH) {
            for (int l = rg; l < LTOK; l += 4) {
                float x = Xn[(size_t)l * CCH + c];
                sum += x;
                sq  += x * x;
            }
        }
        s_sum[rg][c] = sum;
        s_sq[rg][c]  = sq;
        __syncthreads();

        if (threadIdx.x < 64) {
            const int cc = threadIdx.x;
            float S = s_sum[0][cc] + s_sum[1][cc] + s_sum[2][cc] + s_sum[3][cc];
            float Q = s_sq[0][cc]  + s_sq[1][cc]  + s_sq[2][cc]  + s_sq[3][cc];
            float mean = S * (1.0f / (float)LTOK);
            float var  = (Q - (float)LTOK * mean * mean) * (1.0f / (float)(LTOK - 1));
            var = var > 0.0f ? var : 0.0f;
            if (cc < CCH) {
                s_stats[cc]       = mean;
                s_stats[CCH + cc] = sqrtf(var);
            }
        }
        __syncthreads();

        if (threadIdx.x < KCL) {
            const float* cen = centroids + (size_t)threadIdx.x * (2 * CCH);
            float d = 0.0f;
            for (int j = 0; j < 2 * CCH; ++j) {
                float t = s_stats[j] - cen[j];
                d += t * t;
            }
            s_d2[threadIdx.x] = d;
        }
        __syncthreads();

        if (threadIdx.x == 0) {
            int best = 0;
            float bd = s_d2[0];
            for (int k = 1; k < KCL; ++k) {
                if (s_d2[k] < bd) { bd = s_d2[k]; best = k; }
            }
            s_label = best;
            labels[n] = best;    // consumed by the BW gather kernel
        }
        __syncthreads();
    }

    // ---------------- Phase 2: XA = X @ A[label] via WMMA -------------------
    const int lab = s_label;
    const float* Am = A_all + (size_t)lab * CCH * CCH;

    const int lane = threadIdx.x & 31;
    const int wave = threadIdx.x >> 5;
    const int hi   = lane >> 4;
    const int l16  = lane & 15;

    // Stationary B-operand fragments of A[label]: b(k, ncol), k = ks*4 + v + 2*hi
    v2f bfrag[4][15];
#pragma unroll
    for (int nt = 0; nt < 4; ++nt) {
        const int ncol = nt * 16 + l16;
        const bool nok = ncol < CCH;
#pragma unroll
        for (int ks = 0; ks < 15; ++ks) {
            const int k0 = ks * 4 + 2 * hi;
            v2f b;
            b.x = nok ? Am[(size_t)k0 * CCH + ncol] : 0.0f;
            b.y = nok ? Am[(size_t)(k0 + 1) * CCH + ncol] : 0.0f;
            bfrag[nt][ks] = b;
        }
    }

    const v8f vzero = {0.f, 0.f, 0.f, 0.f, 0.f, 0.f, 0.f, 0.f};

    // 64 M-tiles of 16 token-rows; 8 per wave.
    for (int mt = wave; mt < LTOK / 16; mt += 8) {
        const float* Xrow = Xn + (size_t)(mt * 16 + l16) * CCH;
        v2f afrag[15];
#pragma unroll
        for (int ks = 0; ks < 15; ++ks) {
            const int k0 = ks * 4 + 2 * hi;
            v2f a;
            a.x = Xrow[k0];       // two consecutive floats -> global_load_b64
            a.y = Xrow[k0 + 1];
            afrag[ks] = a;
        }

        v8f acc[4];
#pragma unroll
        for (int nt = 0; nt < 4; ++nt) acc[nt] = vzero;

#pragma unroll
        for (int ks = 0; ks < 15; ++ks) {
#pragma unroll
            for (int nt = 0; nt < 4; ++nt) {
                acc[nt] = __builtin_amdgcn_wmma_f32_16x16x4_f32(
                    false, afrag[ks], false, bfrag[nt][ks],
                    (short)0, acc[nt], false, false);
            }
        }

#pragma unroll
        for (int nt = 0; nt < 4; ++nt) {
            const int col = nt * 16 + l16;
            if (col < CCH) {
#pragma unroll
                for (int v = 0; v < 8; ++v) {
                    const int row = mt * 16 + v + 8 * hi;
                    On[(size_t)row * CCH + col] = acc[nt][v];
                }
            }
        }
    }
}

// ---------------------------------------------------------------------------
// BW_k = B_k @ weight for the 8 clusters, into padded [8][64][64].
// One wave32 per cluster; only 8 distinct BW matrices exist (dedup).
// ---------------------------------------------------------------------------
__global__ __launch_bounds__(32) void bw_compute_kernel(
    const float* __restrict__ B_all,   // [K, C, C]
    const float* __restrict__ Wt,      // [C, C]
    float* __restrict__ BWp)           // [K, 64, 64] workspace
{
    const int kk = blockIdx.x;
    const float* Bm = B_all + (size_t)kk * CCH * CCH;
    float* Out = BWp + (size_t)kk * CPAD * CPAD;

    const int lane = threadIdx.x & 31;
    const int hi   = lane >> 4;
    const int l16  = lane & 15;

    // B-operand (weight) fragments: element (k, ncol), k = ks*4 + v + 2*hi
    v2f wf[4][15];
#pragma unroll
    for (int nt = 0; nt < 4; ++nt) {
        const int ncol = nt * 16 + l16;
        const bool nok = ncol < CCH;
#pragma unroll
        for (int ks = 0; ks < 15; ++ks) {
            const int k0 = ks * 4 + 2 * hi;
            v2f b;
            b.x = nok ? Wt[(size_t)k0 * CCH + ncol] : 0.0f;
            b.y = nok ? Wt[(size_t)(k0 + 1) * CCH + ncol] : 0.0f;
            wf[nt][ks] = b;
        }
    }

    const v8f vzero = {0.f, 0.f, 0.f, 0.f, 0.f, 0.f, 0.f, 0.f};

#pragma unroll
    for (int mt = 0; mt < 4; ++mt) {
        const int mrow = mt * 16 + l16;
        const bool mok = mrow < CCH;
        v2f af[15];
#pragma unroll
        for (int ks = 0; ks < 15; ++ks) {
            const int k0 = ks * 4 + 2 * hi;
            v2f a;
            a.x = mok ? Bm[(size_t)mrow * CCH + k0] : 0.0f;
            a.y = mok ? Bm[(size_t)mrow * CCH + k0 + 1] : 0.0f;
            af[ks] = a;
        }
        v8f acc[4];
#pragma unroll
        for (int nt = 0; nt < 4; ++nt) acc[nt] = vzero;
#pragma unroll
        for (int ks = 0; ks < 15; ++ks) {
#pragma unroll
            for (int nt = 0; nt < 4; ++nt) {
                acc[nt] = __builtin_amdgcn_wmma_f32_16x16x4_f32(
                    false, af[ks], false, wf[nt][ks],
                    (short)0, acc[nt], false, false);
            }
        }
#pragma unroll
        for (int nt = 0; nt < 4; ++nt) {
            const int col = nt * 16 + l16;
#pragma unroll
            for (int v = 0; v < 8; ++v) {
                const int row = mt * 16 + v + 8 * hi;
                Out[(size_t)row * CPAD + col] = acc[nt][v];
            }
        }
    }
}

// ---------------------------------------------------------------------------
// Broadcast BW_k -> out per label (pure bandwidth, ~30 MB)
// ---------------------------------------------------------------------------
__global__ __launch_bounds__(256) void gather_bw_kernel(
    const float* __restrict__ BWp,     // [K, 64, 64]
    const int* __restrict__ labels,    // [N]
    float* __restrict__ out)           // [N, 60, 60]
{
    const size_t total = (size_t)NSAMP * CCH * CCH;
    size_t i = (size_t)blockIdx.x * blockDim.x + threadIdx.x;
    if (i >= total) return;
    const int n = (int)(i / (CCH * CCH));
    const int r = (int)(i % (CCH * CCH));
    const int row = r / CCH;
    const int col = r % CCH;
    out[i] = BWp[(size_t)labels[n] * CPAD * CPAD + (size_t)row * CPAD + col];
}

// ---------------------------------------------------------------------------
extern "C" void kernel_launch(void* const* d_in, const int* in_sizes, int n_in,
                              void* d_out, int out_size, void* d_ws, size_t ws_size,
                              hipStream_t stream) {
    (void)in_sizes; (void)n_in; (void)out_size; (void)ws_size;

    const float* X         = (const float*)d_in[0];  // [N, L, C]
    const float* weight    = (const float*)d_in[1];  // [C, C]
    const float* A_all     = (const float*)d_in[2];  // [K, C, C]
    const float* B_all     = (const float*)d_in[3];  // [K, C, C]
    const float* centroids = (const float*)d_in[4];  // [K, 2C]

    float* out_XA = (float*)d_out;                               // [N, L, C]
    float* out_BW = out_XA + (size_t)NSAMP * LTOK * CCH;         // [N, C, C]

    // Workspace layout: labels (int[N]) then padded BW [K][64][64] floats.
    int*   labels = (int*)d_ws;
    float* BWp    = (float*)((char*)d_ws + (((size_t)NSAMP * sizeof(int) + 255) & ~(size_t)255));

    bw_compute_kernel<<<KCL, 32, 0, stream>>>(B_all, weight, BWp);
    fused_stats_xa_kernel<<<NSAMP, 256, 0, stream>>>(X, centroids, A_all, labels, out_XA);

    const size_t total_bw = (size_t)NSAMP * CCH * CCH;
    const int gblocks = (int)((total_bw + 255) / 256);
    gather_bw_kernel<<<gblocks, 256, 0, stream>>>(BWp, labels, out_BW);
}